// LearnableWaveletTransform_31421980738294
// MI455X (gfx1250) — compile-verified
//
#include <hip/hip_runtime.h>

typedef __attribute__((ext_vector_type(2))) float v2f;
typedef __attribute__((ext_vector_type(8))) float v8f;
typedef __attribute__((ext_vector_type(4))) unsigned int u32x4;
typedef __attribute__((ext_vector_type(8))) int i32x8;
typedef __attribute__((ext_vector_type(4))) int i32x4;

#define KW     96
#define PADW   48
#define TIN    4096
#define TOUT   4097
#define ROWS   8192        // 64*128
#define XPLEN  4192        // 48 zeros + 4096 x + 48 zeros; max A index = 3840+351 = 4191
#define WPLEN  128         // 15 zeros + 96 taps + 17 zeros (Toeplitz lookup, idx 0..126 used)
#define NCHUNK 28          // ceil(112/4) K-chunks of the 16x(16..111) band

#if defined(__gfx1250__) && __has_builtin(__builtin_amdgcn_tensor_load_to_lds) && \
    __has_builtin(__builtin_amdgcn_s_wait_tensorcnt)
#define USE_TDM 1
#else
#define USE_TDM 0
#endif

// D = A(16x4 f32) * B(4x16 f32) + C(16x16 f32), wave32.
__device__ __forceinline__ v8f wmma4(v2f a, v2f b, v8f c) {
  return __builtin_amdgcn_wmma_f32_16x16x4_f32(
      /*neg_a=*/false, a, /*neg_b=*/false, b,
      /*c_mod=*/(short)0, c, /*reuse_a=*/false, /*reuse_b=*/false);
}

// Process NT 256-output tiles for one wave; both filters share the A (x) fragments.
template <int NT>
__device__ __forceinline__ void conv_tiles(
    const float* __restrict__ xp, const float* __restrict__ wpl,
    const float* __restrict__ wph, float* __restrict__ outl,
    float* __restrict__ outh, const int* t0s, int n, int hi) {
  const int koff = 2 * hi;           // A: lanes 0-15 hold K=0,1; lanes 16-31 hold K=2,3
  v8f accl[NT];
  v8f acch[NT];
#pragma unroll
  for (int i = 0; i < NT; ++i) { accl[i] = v8f{}; acch[i] = v8f{}; }

  const int abase = 16 * n + koff;   // A row m == lane&15
  const int bbase = 15 + koff - n;   // B[k][n] = w[4a+k-n] via padded wp lookup

#pragma unroll
  for (int a = 0; a < NCHUNK; ++a) {
    const int s = 4 * a;
    v2f bl, bh;                      // B fragments: shared across the NT tiles
    bl[0] = wpl[bbase + s];
    bl[1] = wpl[bbase + s + 1];
    bh[0] = wph[bbase + s];
    bh[1] = wph[bbase + s + 1];
#pragma unroll
    for (int tI = 0; tI < NT; ++tI) {
      // even float index -> 8B-aligned ds_load_b64 (compiler fuses into ds_load_2addr_b64)
      v2f av = *(const v2f*)(xp + t0s[tI] + abase + s);
      accl[tI] = wmma4(av, bl, accl[tI]);
      acch[tI] = wmma4(av, bh, acch[tI]);
    }
  }

  // C/D layout: VGPR j holds (M=j, N=lane) for lanes 0-15, (M=j+8, N=lane-16) for 16-31.
#pragma unroll
  for (int tI = 0; tI < NT; ++tI) {
#pragma unroll
    for (int j = 0; j < 8; ++j) {
      const int t = t0s[tI] + 16 * (j + 8 * hi) + n;   // always < 4096 for tiles 0..15
      outl[t] = accl[tI][j];
      outh[t] = acch[tI][j];
    }
  }
}

__global__ __launch_bounds__(256) void lwt_wmma_kernel(
    const float* __restrict__ x, const float* __restrict__ hw,
    const float* __restrict__ lw, float* __restrict__ out) {
  __shared__ float xp[XPLEN];
  __shared__ float wpl[WPLEN];
  __shared__ float wph[WPLEN];

  const int tid = threadIdx.x;
  const int row = blockIdx.x;
  const float* __restrict__ xr = x + (size_t)row * TIN;
  const int wave = tid >> 5;   // 8 waves per block
  const int lane = tid & 31;

  // Zero pads (disjoint from the TDM/body region, ordered by the barrier below).
  if (tid < PADW) {
    xp[tid] = 0.f;
    xp[PADW + TIN + tid] = 0.f;
  }

#if USE_TDM
  // Tensor Data Mover: one DMA of the 16 KB row straight into LDS (no VGPR staging).
  // 2-D descriptor: 1 row x 4096 elements of 4 bytes. Wave 0 issues; TENSORcnt
  // wait + the workgroup barrier below order it for all waves.
  if (wave == 0) {
    unsigned lds_addr = (unsigned)(uintptr_t)(xp + PADW);   // LDS aperture: low 32 bits
    unsigned long long ga = (unsigned long long)(uintptr_t)xr;
    u32x4 g0;
    g0[0] = 1u;                                   // count=1 (valid), no gather, user mode
    g0[1] = lds_addr;                             // lds_addr[31:0]
    g0[2] = (unsigned)ga;                         // global_addr[31:0]
    g0[3] = ((unsigned)(ga >> 32) & 0x01FFFFFFu)  // global_addr[56:32]
            | (2u << 30);                         // type=2 ("image")
    i32x8 g1;
    g1[0] = 0x00020000;                           // workgroup_mask=0, data_size=2 (4B)
    g1[1] = (int)((TIN & 0xFFFF) << 16);          // tensor_dim0[15:0] at bits 63:48
    g1[2] = (int)((TIN >> 16) | (1 << 16));       // tensor_dim0[31:16], tensor_dim1=1
    g1[3] = (int)(TIN << 16);                     // tensor_dim1 hi=0, tile_dim0=4096
    g1[4] = 1;                                    // tile_dim1=1, tile_dim2=0
    g1[5] = TIN;                                  // tensor_dim0_stride[31:0]
    g1[6] = 0;                                    // stride hi, tensor_dim1_stride lo
    g1[7] = 0;                                    // tensor_dim1_stride hi
    i32x4 g2 = {0, 0, 0, 0};
    i32x4 g3 = {0, 0, 0, 0};
    i32x8 g4 = {0, 0, 0, 0, 0, 0, 0, 0};          // extra group (6-arg builtin form)
    __builtin_amdgcn_tensor_load_to_lds(g0, g1, g2, g3, g4, /*cpol=*/0);
    __builtin_amdgcn_s_wait_tensorcnt(0);
  }
#else
  // Fallback: 4096 floats = 1024 float4 (xp+48 is 16B aligned).
  {
    const float4* __restrict__ x4 = (const float4*)xr;
    float4* xp4 = (float4*)(xp + PADW);
#pragma unroll 4
    for (int i = tid; i < TIN / 4; i += 256) xp4[i] = x4[i];
  }
#endif

  // Padded Toeplitz filter tables. NOTE output order is (l, h); inputs are (x, h_w, l_w).
  if (tid < WPLEN) {
    wpl[tid] = (tid >= 15 && tid < 15 + KW) ? lw[tid - 15] : 0.f;
  } else {
    const int j = tid - WPLEN;
    wph[j] = (j >= 15 && j < 15 + KW) ? hw[j - 15] : 0.f;
  }
  __syncthreads();

  float* __restrict__ outl = out + (size_t)row * TOUT;
  float* __restrict__ outh = out + (size_t)ROWS * TOUT + (size_t)row * TOUT;

  const int n = lane & 15;
  const int hi = lane >> 4;

  // 16 tiles of 256 outputs cover t in [0, 4096); wave w owns tiles w and w+8.
  int t0s[2] = {256 * wave, 256 * (wave + 8)};
  conv_tiles<2>(xp, wpl, wph, outl, outh, t0s, n, hi);

  // Tail output t = 4096 (one element per filter): tiny scalar dot from LDS.
  if (tid < 2) {
    const float* wp = tid ? wph : wpl;
    float* o = tid ? outh : outl;
    float s = 0.f;
#pragma unroll
    for (int k = 0; k < KW; ++k) s += wp[15 + k] * xp[TIN + k];
    o[TIN] = s;
  }
}

extern "C" void kernel_launch(void* const* d_in, const int* in_sizes, int n_in,
                              void* d_out, int out_size, void* d_ws, size_t ws_size,
                              hipStream_t stream) {
  (void)in_sizes; (void)n_in; (void)out_size; (void)d_ws; (void)ws_size;
  const float* x  = (const float*)d_in[0];
  const float* hw = (const float*)d_in[1];   // h_w
  const float* lw = (const float*)d_in[2];   // l_w
  float* out = (float*)d_out;                // [l_outs | h_outs], each ROWS*TOUT fp32

  lwt_wmma_kernel<<<dim3(ROWS), dim3(256), 0, stream>>>(x, hw, lw, out);
}